// CrossAttentionLayer_63187558859331
// MI455X (gfx1250) — compile-verified
//
#include <hip/hip_runtime.h>
#include <hip/hip_bf16.h>
#include <stddef.h>

// ---------------------------------------------------------------------------
// CDNA5 (gfx1250) ragged cross-attention + MLP layer, fp32 WMMA throughout.
//  - attention: 1 block per (graph, head); strip code templated on TS=Ls/16
//    so all WMMA loops are static (no guards, no phi copies).
//  - GEMM1/2: 128x256 block tiles, K chunked via LDS; B-frags batch-loaded
//    so WMMAs issue back-to-back under a single ds-wait.
//  - BN stats: deterministic 2-stage column reduction (graph-replay safe).
// ---------------------------------------------------------------------------

typedef __attribute__((ext_vector_type(2))) float v2f;
typedef __attribute__((ext_vector_type(8))) float v8f;

static __device__ __forceinline__ v8f wmma_f32(v2f a, v2f b, v8f c) {
  // D = A(16x4) * B(4x16) + C(16x16), fp32 WMMA
  return __builtin_amdgcn_wmma_f32_16x16x4_f32(false, a, false, b, (short)0, c,
                                               false, false);
}

#define HD 256
#define INV_SQRT32 0.17677669529663687f

// ---------------------------------------------------------------------------
// One 16-row dst strip of attention for a (graph, head), TS = Ls/16 tiles.
// Q: [16][34] LDS strip, Ks: [Ls][34] LDS K/V slice, P: [16][132] LDS scratch.
// ---------------------------------------------------------------------------
template <int TS>
__device__ __forceinline__ void attn_strip(const float* __restrict__ Q,
                                           const float* __restrict__ Ks,
                                           float* __restrict__ P,
                                           float* __restrict__ h_msg,
                                           size_t rowbase, int hh, int li,
                                           int half) {
  const v8f vzero = {0.f, 0.f, 0.f, 0.f, 0.f, 0.f, 0.f, 0.f};

  // --- scores: 16 x (TS*16) strip, K=32 -> 8 k-steps ---
  v8f sc[TS];
#pragma unroll
  for (int i = 0; i < TS; ++i) sc[i] = vzero;
#pragma unroll
  for (int ks = 0; ks < 8; ++ks) {
    const int k0 = ks * 4 + 2 * half;
    v2f a = *(const v2f*)&Q[li * 34 + k0];
    v2f bf[TS];
#pragma unroll
    for (int t2 = 0; t2 < TS; ++t2)
      bf[t2] = *(const v2f*)&Ks[(t2 * 16 + li) * 34 + k0];
#pragma unroll
    for (int t2 = 0; t2 < TS; ++t2) sc[t2] = wmma_f32(a, bf[t2], sc[t2]);
  }

  // --- row softmax (row = r + 8*half lives in one half-wave) ---
  float mx[8], sm[8];
#pragma unroll
  for (int r = 0; r < 8; ++r) {
    float m = -3.0e38f;
#pragma unroll
    for (int t2 = 0; t2 < TS; ++t2) m = fmaxf(m, sc[t2][r]);
#pragma unroll
    for (int off = 8; off >= 1; off >>= 1)
      m = fmaxf(m, __shfl_xor(m, off, 16));
    mx[r] = m;
  }
#pragma unroll
  for (int t2 = 0; t2 < TS; ++t2) {
#pragma unroll
    for (int r = 0; r < 8; ++r)
      sc[t2][r] = __expf((sc[t2][r] - mx[r]) * INV_SQRT32);
  }
#pragma unroll
  for (int r = 0; r < 8; ++r) {
    float s = 0.f;
#pragma unroll
    for (int t2 = 0; t2 < TS; ++t2) s += sc[t2][r];
#pragma unroll
    for (int off = 8; off >= 1; off >>= 1) s += __shfl_xor(s, off, 16);
    sm[r] = 1.0f / s;
  }
  // P strip -> LDS (C-layout to row-major, stride 132)
#pragma unroll
  for (int t2 = 0; t2 < TS; ++t2) {
#pragma unroll
    for (int r = 0; r < 8; ++r)
      P[(r + 8 * half) * 132 + t2 * 16 + li] = sc[t2][r] * sm[r];
  }

  // --- out = P (16 x Ls) @ V (Ls x 32) ---
  v8f ao0 = vzero, ao1 = vzero;
#pragma unroll
  for (int km = 0; km < TS * 4; ++km) {
    const int k0 = km * 4 + 2 * half;
    v2f a = *(const v2f*)&P[li * 132 + k0];
    v2f b0, b1;
    b0.x = Ks[(k0 + 0) * 34 + li];
    b0.y = Ks[(k0 + 1) * 34 + li];
    b1.x = Ks[(k0 + 0) * 34 + 16 + li];
    b1.y = Ks[(k0 + 1) * 34 + 16 + li];
    ao0 = wmma_f32(a, b0, ao0);
    ao1 = wmma_f32(a, b1, ao1);
  }
#pragma unroll
  for (int r = 0; r < 8; ++r) {
    const size_t row = rowbase + r + 8 * half;
    h_msg[row * HD + li * 8 + hh] = ao0[r];
    h_msg[row * HD + (16 + li) * 8 + hh] = ao1[r];
  }
}

// ---------------------------------------------------------------------------
// Attention: grid = (num_graphs, 8 heads), block = 128 threads (4 waves).
// ---------------------------------------------------------------------------
__global__ __launch_bounds__(128) void attn_kernel(
    const float* __restrict__ src_h, const float* __restrict__ dst_h,
    const int* __restrict__ slens, const int* __restrict__ dlens,
    float* __restrict__ h_msg) {
  __shared__ __align__(16) float Ks[128 * 34];     // K/V head slice
  __shared__ __align__(16) float Qs[4][16 * 34];   // per-wave Q strip
  __shared__ __align__(16) float Ps[4][16 * 132];  // per-wave P strip
  __shared__ long long red[128];
  __shared__ int meta[4];

  const int g = blockIdx.x, hh = blockIdx.y;
  const int t = threadIdx.x;
  const int wave = t >> 5, lane = t & 31, li = lane & 15, half = lane >> 4;

  // --- row offsets for this graph (packed dual prefix sum) ---
  long long p = 0;
  for (int i = t; i < g; i += 128)
    p += (long long)(unsigned)slens[i] + ((long long)(unsigned)dlens[i] << 32);
  red[t] = p;
  __syncthreads();
  for (int s = 64; s > 0; s >>= 1) {
    if (t < s) red[t] += red[t + s];
    __syncthreads();
  }
  if (t == 0) {
    meta[0] = (int)(red[0] & 0xffffffffLL);
    meta[1] = (int)(red[0] >> 32);
    meta[2] = slens[g];
    meta[3] = dlens[g];
  }
  __syncthreads();
  const int offs = meta[0], offd = meta[1], Ls = meta[2], Ld = meta[3];

  // --- stage K/V head slice: Ks[m][d] = src[offs+m][d*8+h] ---
  for (int idx = t; idx < Ls * 32; idx += 128) {
    int m = idx >> 5, d = idx & 31;
    Ks[m * 34 + d] = src_h[(size_t)(offs + m) * HD + d * 8 + hh];
  }
  __syncthreads();

  const int Ts = Ls >> 4, Td = Ld >> 4;

#pragma unroll 1
  for (int strip = wave; strip < Td; strip += 4) {
    const int n0 = strip * 16;
    float* Q = Qs[wave];
    float* P = Ps[wave];
    // Q strip: Q[r][d] = dst[offd+n0+r][d*8+h]  (wave-local, no barrier)
    for (int idx = lane; idx < 512; idx += 32) {
      int r = idx >> 5, d = idx & 31;
      Q[r * 34 + d] = dst_h[(size_t)(offd + n0 + r) * HD + d * 8 + hh];
    }
    const size_t rowbase = (size_t)(offd + n0);
    switch (Ts) {
      case 1: attn_strip<1>(Q, Ks, P, h_msg, rowbase, hh, li, half); break;
      case 2: attn_strip<2>(Q, Ks, P, h_msg, rowbase, hh, li, half); break;
      case 3: attn_strip<3>(Q, Ks, P, h_msg, rowbase, hh, li, half); break;
      case 4: attn_strip<4>(Q, Ks, P, h_msg, rowbase, hh, li, half); break;
      case 5: attn_strip<5>(Q, Ks, P, h_msg, rowbase, hh, li, half); break;
      case 6: attn_strip<6>(Q, Ks, P, h_msg, rowbase, hh, li, half); break;
      case 7: attn_strip<7>(Q, Ks, P, h_msg, rowbase, hh, li, half); break;
      default: attn_strip<8>(Q, Ks, P, h_msg, rowbase, hh, li, half); break;
    }
  }
}

// ---------------------------------------------------------------------------
// GEMM: Y[n][c] = sum_k X[n][k] * W[c][k] + bias[c]
// KDIM=512: X = [X0 | X1].  FUSE: X = relu(X0*scale + shift) (BN1+ReLU).
// Block: 256 threads (8 waves), tile 128 rows x 256 cols, K chunked by 32.
// ---------------------------------------------------------------------------
template <int KDIM, bool FUSE>
__global__ __launch_bounds__(256) void gemm_kernel(
    const float* __restrict__ X0, const float* __restrict__ X1,
    const float* __restrict__ W, const float* __restrict__ bias,
    const float* __restrict__ scaleIn, const float* __restrict__ shiftIn,
    float* __restrict__ Y) {
  __shared__ __align__(16) float Xs[128 * 36];
  __shared__ __align__(16) float Ws[256 * 36];
  const int t = threadIdx.x;
  const int wave = t >> 5, lane = t & 31, li = lane & 15, half = lane >> 4;
  const int row0 = blockIdx.x * 128;
  const v8f vzero = {0.f, 0.f, 0.f, 0.f, 0.f, 0.f, 0.f, 0.f};
  v8f acc[16];
#pragma unroll
  for (int i = 0; i < 16; ++i) acc[i] = vzero;

  const int xrow = t >> 1, xh = t & 1;
#pragma unroll 1
  for (int kc = 0; kc < KDIM / 32; ++kc) {
    // stage X chunk [128 rows x 32 k]
    {
      const float* xb = X0;
      int cc = kc * 32;
      if (KDIM == 512 && kc >= 8) { xb = X1; cc -= 256; }
      const float* xsrc = xb + (size_t)(row0 + xrow) * HD + cc + xh * 16;
#pragma unroll
      for (int i = 0; i < 4; ++i) {
        float4 v = *(const float4*)(xsrc + i * 4);
        if (FUSE) {
          const int c = cc + xh * 16 + i * 4;
          v.x = fmaxf(fmaf(v.x, scaleIn[c + 0], shiftIn[c + 0]), 0.f);
          v.y = fmaxf(fmaf(v.y, scaleIn[c + 1], shiftIn[c + 1]), 0.f);
          v.z = fmaxf(fmaf(v.z, scaleIn[c + 2], shiftIn[c + 2]), 0.f);
          v.w = fmaxf(fmaf(v.w, scaleIn[c + 3], shiftIn[c + 3]), 0.f);
        }
        *(float4*)&Xs[xrow * 36 + xh * 16 + i * 4] = v;
      }
    }
    // stage W chunk [256 cols x 32 k]
    {
      const float* wsrc = W + (size_t)t * KDIM + kc * 32;
#pragma unroll
      for (int i = 0; i < 8; ++i)
        *(float4*)&Ws[t * 36 + i * 4] = *(const float4*)(wsrc + i * 4);
    }
    __syncthreads();

#pragma unroll
    for (int ks = 0; ks < 8; ++ks) {
      const int k0 = ks * 4 + 2 * half;
      v2f a = *(const v2f*)&Xs[(wave * 16 + li) * 36 + k0];
      v2f bf[16];
#pragma unroll
      for (int ct = 0; ct < 16; ++ct)
        bf[ct] = *(const v2f*)&Ws[(ct * 16 + li) * 36 + k0];
#pragma unroll
      for (int ct = 0; ct < 16; ++ct) acc[ct] = wmma_f32(a, bf[ct], acc[ct]);
    }
    __syncthreads();
  }

  // epilogue: bias add, store (in-place safe: block reads only its own rows)
#pragma unroll
  for (int ct = 0; ct < 16; ++ct) {
    const int c = ct * 16 + li;
    const float bv = bias[c];
#pragma unroll
    for (int r = 0; r < 8; ++r) {
      const size_t row = (size_t)(row0 + wave * 16 + r + 8 * half);
      Y[row * HD + c] = acc[ct][r] + bv;
    }
  }
}

// ---------------------------------------------------------------------------
// BN stats: deterministic 2-stage column reduction over N rows x 256 cols.
// ---------------------------------------------------------------------------
__global__ __launch_bounds__(256) void reduce_cols(const float* __restrict__ Y,
                                                   float* __restrict__ partials,
                                                   int rows_per_block) {
  const int t = threadIdx.x, b = blockIdx.x;
  const size_t base = (size_t)b * rows_per_block * HD;
  float s = 0.f, s2 = 0.f;
  for (int i = 0; i < rows_per_block; ++i) {
    float v = Y[base + (size_t)i * HD + t];
    s += v;
    s2 += v * v;
  }
  partials[b * 512 + t] = s;
  partials[b * 512 + 256 + t] = s2;
}

__global__ __launch_bounds__(256) void finalize_stats(
    const float* __restrict__ partials, const float* __restrict__ gamma,
    const float* __restrict__ beta, float* __restrict__ stats, float inv_n) {
  const int t = threadIdx.x;
  float s = 0.f, s2 = 0.f;
  for (int i = 0; i < 256; ++i) {
    s += partials[i * 512 + t];
    s2 += partials[i * 512 + 256 + t];
  }
  const float mean = s * inv_n;
  const float var = s2 * inv_n - mean * mean;
  const float sc = gamma[t] * rsqrtf(var + 1e-5f);
  stats[t] = sc;                         // scale
  stats[256 + t] = beta[t] - mean * sc;  // shift
}

// out = h + y2*scale2[c] + shift2[c]
__global__ __launch_bounds__(256) void final_add(const float* __restrict__ H,
                                                 const float* __restrict__ Y,
                                                 const float* __restrict__ st,
                                                 float* __restrict__ out) {
  const size_t i = (size_t)blockIdx.x * blockDim.x + threadIdx.x;  // float4 idx
  const int c = (int)(i & 63) * 4;
  float4 hv = ((const float4*)H)[i];
  float4 yv = ((const float4*)Y)[i];
  float4 o;
  o.x = hv.x + yv.x * st[c + 0] + st[256 + c + 0];
  o.y = hv.y + yv.y * st[c + 1] + st[256 + c + 1];
  o.z = hv.z + yv.z * st[c + 2] + st[256 + c + 2];
  o.w = hv.w + yv.w * st[c + 3] + st[256 + c + 3];
  ((float4*)out)[i] = o;
}

// ---------------------------------------------------------------------------
extern "C" void kernel_launch(void* const* d_in, const int* in_sizes, int n_in,
                              void* d_out, int out_size, void* d_ws,
                              size_t ws_size, hipStream_t stream) {
  const float* src_h = (const float*)d_in[0];
  const float* dst_h = (const float*)d_in[1];
  const int* slens = (const int*)d_in[2];
  const int* dlens = (const int*)d_in[3];
  const float* W1 = (const float*)d_in[4];
  const float* b1 = (const float*)d_in[5];
  const float* g1 = (const float*)d_in[6];
  const float* be1 = (const float*)d_in[7];
  const float* W2 = (const float*)d_in[8];
  const float* b2 = (const float*)d_in[9];
  const float* g2 = (const float*)d_in[10];
  const float* be2 = (const float*)d_in[11];
  float* out = (float*)d_out;

  const int N = in_sizes[0] / HD;  // 40960 rows
  const int G = in_sizes[2];       // 512 graphs

  float* ws = (float*)d_ws;
  float* h_msg = ws;                     // N*256
  float* y = h_msg + (size_t)N * HD;     // N*256 (y1, then y2 in place)
  float* partials = y + (size_t)N * HD;  // 256*512
  float* stats1 = partials + 256 * 512;  // 512
  float* stats2 = stats1 + 512;          // 512

  // 1) per-(graph, head) attention -> h_msg
  attn_kernel<<<dim3(G, 8), 128, 0, stream>>>(src_h, dst_h, slens, dlens,
                                              h_msg);
  // 2) y1 = [h | h_msg] @ W1^T + b1
  gemm_kernel<512, false><<<N / 128, 256, 0, stream>>>(dst_h, h_msg, W1, b1,
                                                       nullptr, nullptr, y);
  // 3) BN1 stats
  reduce_cols<<<256, 256, 0, stream>>>(y, partials, N / 256);
  finalize_stats<<<1, 256, 0, stream>>>(partials, g1, be1, stats1,
                                        1.0f / (float)N);
  // 4) y2 = relu(bn1(y1)) @ W2^T + b2   (BN1+ReLU fused into X load, in place)
  gemm_kernel<256, true><<<N / 128, 256, 0, stream>>>(y, nullptr, W2, b2,
                                                      stats1, stats1 + 256, y);
  // 5) BN2 stats
  reduce_cols<<<256, 256, 0, stream>>>(y, partials, N / 256);
  finalize_stats<<<1, 256, 0, stream>>>(partials, g2, be2, stats2,
                                        1.0f / (float)N);
  // 6) out = h + bn2(y2)
  final_add<<<N / 4, 256, 0, stream>>>(dst_h, y, stats2, out);
}